// CryptoGNN_46677704573771
// MI455X (gfx1250) — compile-verified
//
#include <hip/hip_runtime.h>

#define N_NODES   50000
#define N_EDGES   1600000
#define N_FEAT    8
#define HID       64
#define N_GRAPHS  512
#define N_CLASSES 3
#define BN_EPS    1e-5f
#define ASTR      68   // padded LDS row stride for 16x64 A tiles (even -> 8B aligned pairs)

typedef __attribute__((ext_vector_type(2))) float v2f;
typedef __attribute__((ext_vector_type(8))) float v8f;

// D = A(16x4,f32) * B(4x16,f32) + C(16x16,f32), wave32 collaborative.
__device__ inline v8f wmma4(v2f a, v2f b, v8f c) {
  return __builtin_amdgcn_wmma_f32_16x16x4_f32(
      /*neg_a=*/false, a, /*neg_b=*/false, b,
      /*c_mod=*/(short)0, c, /*reuse_a=*/false, /*reuse_b=*/false);
}

// ---------------- workspace layout (float offsets) ----------------
// accumulators (must be zeroed every launch) are contiguous at the front
#define OFF_AGG    0                         // 50000*8
#define OFF_DEG    400000                    // 50000
#define OFF_NBSUM  450000                    // 50000*64
#define OFF_GSUM   3650000                   // 512*64
#define OFF_GCNT   3682768                   // 512
#define OFF_S1     3683280                   // 64
#define OFF_Q1     3683344                   // 64
#define OFF_S2     3683408                   // 64
#define OFF_Q2     3683472                   // 64
#define ZERO_COUNT 3683536
#define OFF_SCALE1 3683536                   // 64
#define OFF_SHIFT1 3683600                   // 64
#define OFF_SCALE2 3683664                   // 64
#define OFF_SHIFT2 3683728                   // 64
#define OFF_H      3683792                   // 50000*64  (16B aligned)
#define OFF_H3     6883792                   // 50000*64

__global__ void k_zero(float* __restrict__ p, int n) {
  int i = blockIdx.x * blockDim.x + threadIdx.x;
  if (i < n) p[i] = 0.0f;
}

// agg[dst] += x[src] (8 feats), deg[dst] += 1
__global__ void k_gin_agg(const float* __restrict__ x, const int* __restrict__ ei,
                          float* __restrict__ agg, float* __restrict__ deg) {
  int e = blockIdx.x * blockDim.x + threadIdx.x;
  if (e >= N_EDGES) return;
  int s = ei[e];
  int d = ei[N_EDGES + e];
  const float4* xr = (const float4*)(x + (size_t)s * 8);
  float4 a = xr[0], b = xr[1];
  float* ad = agg + (size_t)d * 8;
  atomicAdd(ad + 0, a.x); atomicAdd(ad + 1, a.y);
  atomicAdd(ad + 2, a.z); atomicAdd(ad + 3, a.w);
  atomicAdd(ad + 4, b.x); atomicAdd(ad + 5, b.y);
  atomicAdd(ad + 6, b.z); atomicAdd(ad + 7, b.w);
  atomicAdd(deg + d, 1.0f);
}

// Load a 64x64 weight (row-major [k][c]) into LDS in K-pair interleaved form:
//   wp[(p*64 + c)*2 + hk] = w[(2p + hk)*64 + c]
// so the WMMA B operand (k0,k0+1 at column c) is one aligned 8-byte LDS load.
__device__ inline void load_wpair(float* __restrict__ wp, const float* __restrict__ w,
                                  int tid, int nthreads) {
  for (int i = tid; i < 64 * 64; i += nthreads) {
    int p  = i >> 7;        // k-pair index 0..31
    int r  = i & 127;
    int c  = r >> 1;        // column 0..63
    int hk = r & 1;         // which k of the pair
    wp[i] = w[(2 * p + hk) * 64 + c];
  }
}

// h = relu((x+agg)@w1 + b1) @ w2 + b2 ; also accumulate bn1 sum/sumsq.
// 128 threads = 4 waves; each wave computes a 16-node x 64-col tile with WMMA.
__global__ __launch_bounds__(128) void k_gin_mlp(
    const float* __restrict__ x, const float* __restrict__ agg,
    const float* __restrict__ w1, const float* __restrict__ b1,
    const float* __restrict__ w2, const float* __restrict__ b2,
    float* __restrict__ h, float* __restrict__ s1, float* __restrict__ q1) {
  __shared__ float w1s[8 * 64];
  __shared__ float b1s[64];
  __shared__ __align__(16) float w2p[64 * 64];  // K-pair interleaved
  __shared__ float b2s[64];
  __shared__ float xin[4][16 * 8];
  __shared__ __align__(16) float at[4][16 * ASTR];

  const int tid = threadIdx.x;
  for (int i = tid; i < 8 * 64; i += 128) w1s[i] = w1[i];
  for (int i = tid; i < 64; i += 128) { b1s[i] = b1[i]; b2s[i] = b2[i]; }
  load_wpair(w2p, w2, tid, 128);

  const int wv = tid >> 5;
  const int lane = tid & 31;
  const int lhalf = lane >> 4;
  const int lm = lane & 15;
  const int nodeBase = blockIdx.x * 64 + wv * 16;

  // stage x + agg for this wave's 16 nodes (clamped for tail)
  for (int i = lane; i < 16 * 8; i += 32) {
    int rr = i >> 3, k = i & 7;
    int node = nodeBase + rr; if (node >= N_NODES) node = N_NODES - 1;
    xin[wv][i] = x[(size_t)node * 8 + k] + agg[(size_t)node * 8 + k];
  }
  __syncthreads();

  // layer 1 (8 -> 64), scalar FMAs, result (post-ReLU) into LDS A-tile
  for (int i = lane; i < 16 * 64; i += 32) {
    int rr = i >> 6, c = i & 63;
    float s = b1s[c];
#pragma unroll
    for (int k = 0; k < 8; ++k) s += xin[wv][rr * 8 + k] * w1s[k * 64 + c];
    at[wv][rr * ASTR + c] = fmaxf(s, 0.0f);
  }
  __syncthreads();

  // layer 2 (64 -> 64) via v_wmma_f32_16x16x4_f32
  v8f acc[4];
#pragma unroll
  for (int t = 0; t < 4; ++t)
#pragma unroll
    for (int r = 0; r < 8; ++r) acc[t][r] = 0.0f;

#pragma unroll
  for (int kc = 0; kc < 16; ++kc) {
    const int k0 = kc * 4 + lhalf * 2;       // even -> 8B aligned
    const int p  = kc * 2 + lhalf;           // k-pair index
    const v2f a = *(const v2f*)&at[wv][lm * ASTR + k0];
#pragma unroll
    for (int t = 0; t < 4; ++t) {
      const v2f b = *(const v2f*)&w2p[(p * 64 + t * 16 + lm) * 2];
      acc[t] = wmma4(a, b, acc[t]);
    }
  }

  // write h (+bias) and accumulate bn1 statistics
#pragma unroll
  for (int t = 0; t < 4; ++t) {
    const int col = t * 16 + lm;
    const float bias = b2s[col];
    float ps = 0.0f, pq = 0.0f;
#pragma unroll
    for (int r = 0; r < 8; ++r) {
      int node = nodeBase + r + lhalf * 8;
      float v = acc[t][r] + bias;
      if (node < N_NODES) {
        h[(size_t)node * 64 + col] = v;
        ps += v; pq += v * v;
      }
    }
    ps += __shfl_xor(ps, 16, 32);
    pq += __shfl_xor(pq, 16, 32);
    if (lhalf == 0) { atomicAdd(&s1[col], ps); atomicAdd(&q1[col], pq); }
  }
}

__global__ void k_bn_finalize(const float* __restrict__ s, const float* __restrict__ q,
                              const float* __restrict__ g, const float* __restrict__ b,
                              float* __restrict__ scale, float* __restrict__ shift) {
  int c = threadIdx.x;
  if (c < HID) {
    float inv = 1.0f / (float)N_NODES;
    float mu = s[c] * inv;
    float var = q[c] * inv - mu * mu;
    float sc = g[c] * rsqrtf(var + BN_EPS);
    scale[c] = sc;
    shift[c] = b[c] - mu * sc;
  }
}

__global__ void k_bn_relu(float* __restrict__ h, const float* __restrict__ scale,
                          const float* __restrict__ shift) {
  int i = blockIdx.x * blockDim.x + threadIdx.x;
  if (i < N_NODES * HID) {
    int c = i & 63;
    h[i] = fmaxf(h[i] * scale[c] + shift[c], 0.0f);
  }
}

// nbsum[dst] += h[src] (64 feats); 16 threads per edge, float4 each
__global__ void k_sage_agg(const float* __restrict__ h, const int* __restrict__ ei,
                           float* __restrict__ nbsum) {
  int t = blockIdx.x * blockDim.x + threadIdx.x;
  if (t >= N_EDGES * 16) return;
  int e = t >> 4, q = (t & 15) * 4;
  int s = ei[e], d = ei[N_EDGES + e];
  const float4 v = *(const float4*)(h + (size_t)s * 64 + q);
  float* o = nbsum + (size_t)d * 64 + q;
  atomicAdd(o + 0, v.x); atomicAdd(o + 1, v.y);
  atomicAdd(o + 2, v.z); atomicAdd(o + 3, v.w);
}

// h3 = nb_mean@wl + bl + h@wr ; accumulate bn2 stats. Two WMMA GEMMs into one acc.
__global__ __launch_bounds__(128) void k_sage_lin(
    const float* __restrict__ h, const float* __restrict__ nbsum,
    const float* __restrict__ deg,
    const float* __restrict__ wl, const float* __restrict__ bl,
    const float* __restrict__ wr,
    float* __restrict__ h3, float* __restrict__ s2, float* __restrict__ q2) {
  __shared__ __align__(16) float wlp[64 * 64];  // K-pair interleaved
  __shared__ __align__(16) float wrp[64 * 64];  // K-pair interleaved
  __shared__ float bls[64];
  __shared__ __align__(16) float at[4][16 * ASTR];

  const int tid = threadIdx.x;
  load_wpair(wlp, wl, tid, 128);
  load_wpair(wrp, wr, tid, 128);
  for (int i = tid; i < 64; i += 128) bls[i] = bl[i];

  const int wv = tid >> 5;
  const int lane = tid & 31;
  const int lhalf = lane >> 4;
  const int lm = lane & 15;
  const int nodeBase = blockIdx.x * 64 + wv * 16;

  v8f acc[4];
#pragma unroll
  for (int t = 0; t < 4; ++t)
#pragma unroll
    for (int r = 0; r < 8; ++r) acc[t][r] = 0.0f;

  // pass 1: A = nb_mean
  for (int i = lane; i < 16 * 64; i += 32) {
    int rr = i >> 6, c = i & 63;
    int node = nodeBase + rr; if (node >= N_NODES) node = N_NODES - 1;
    float dg = fmaxf(deg[node], 1.0f);
    at[wv][rr * ASTR + c] = nbsum[(size_t)node * 64 + c] / dg;
  }
  __syncthreads();
#pragma unroll
  for (int kc = 0; kc < 16; ++kc) {
    const int k0 = kc * 4 + lhalf * 2;
    const int p  = kc * 2 + lhalf;
    const v2f a = *(const v2f*)&at[wv][lm * ASTR + k0];
#pragma unroll
    for (int t = 0; t < 4; ++t) {
      const v2f b = *(const v2f*)&wlp[(p * 64 + t * 16 + lm) * 2];
      acc[t] = wmma4(a, b, acc[t]);
    }
  }
  __syncthreads();

  // pass 2: A = h
  for (int i = lane; i < 16 * 64; i += 32) {
    int rr = i >> 6, c = i & 63;
    int node = nodeBase + rr; if (node >= N_NODES) node = N_NODES - 1;
    at[wv][rr * ASTR + c] = h[(size_t)node * 64 + c];
  }
  __syncthreads();
#pragma unroll
  for (int kc = 0; kc < 16; ++kc) {
    const int k0 = kc * 4 + lhalf * 2;
    const int p  = kc * 2 + lhalf;
    const v2f a = *(const v2f*)&at[wv][lm * ASTR + k0];
#pragma unroll
    for (int t = 0; t < 4; ++t) {
      const v2f b = *(const v2f*)&wrp[(p * 64 + t * 16 + lm) * 2];
      acc[t] = wmma4(a, b, acc[t]);
    }
  }

#pragma unroll
  for (int t = 0; t < 4; ++t) {
    const int col = t * 16 + lm;
    const float bias = bls[col];
    float ps = 0.0f, pq = 0.0f;
#pragma unroll
    for (int r = 0; r < 8; ++r) {
      int node = nodeBase + r + lhalf * 8;
      float v = acc[t][r] + bias;
      if (node < N_NODES) {
        h3[(size_t)node * 64 + col] = v;
        ps += v; pq += v * v;
      }
    }
    ps += __shfl_xor(ps, 16, 32);
    pq += __shfl_xor(pq, 16, 32);
    if (lhalf == 0) { atomicAdd(&s2[col], ps); atomicAdd(&q2[col], pq); }
  }
}

// bn2 + relu + global mean-pool accumulation
__global__ void k_bn2_pool(const float* __restrict__ h3, const float* __restrict__ scale,
                           const float* __restrict__ shift, const int* __restrict__ batch,
                           float* __restrict__ gsum, float* __restrict__ gcnt) {
  int t = blockIdx.x * blockDim.x + threadIdx.x;
  if (t >= N_NODES * 16) return;
  int node = t >> 4, q = (t & 15) * 4;
  int g = batch[node];
  const float4 v = *(const float4*)(h3 + (size_t)node * 64 + q);
  float rx = fmaxf(v.x * scale[q + 0] + shift[q + 0], 0.0f);
  float ry = fmaxf(v.y * scale[q + 1] + shift[q + 1], 0.0f);
  float rz = fmaxf(v.z * scale[q + 2] + shift[q + 2], 0.0f);
  float rw = fmaxf(v.w * scale[q + 3] + shift[q + 3], 0.0f);
  float* o = gsum + (size_t)g * 64 + q;
  atomicAdd(o + 0, rx); atomicAdd(o + 1, ry);
  atomicAdd(o + 2, rz); atomicAdd(o + 3, rw);
  if (q == 0) atomicAdd(&gcnt[g], 1.0f);
}

// per-graph classifier: relu(mean @ w1 + b1) @ w2 + b2
__global__ __launch_bounds__(32) void k_classifier(
    const float* __restrict__ gsum, const float* __restrict__ gcnt,
    const float* __restrict__ w1, const float* __restrict__ b1,
    const float* __restrict__ w2, const float* __restrict__ b2,
    float* __restrict__ out) {
  __shared__ float gm[64];
  __shared__ float hid[32];
  int g = blockIdx.x, t = threadIdx.x;
  float cnt = fmaxf(gcnt[g], 1.0f);
  for (int i = t; i < 64; i += 32) gm[i] = gsum[(size_t)g * 64 + i] / cnt;
  __syncthreads();
  float s = b1[t];
#pragma unroll 8
  for (int k = 0; k < 64; ++k) s += gm[k] * w1[k * 32 + t];
  hid[t] = fmaxf(s, 0.0f);
  __syncthreads();
  if (t < N_CLASSES) {
    float o = b2[t];
#pragma unroll
    for (int k = 0; k < 32; ++k) o += hid[k] * w2[k * N_CLASSES + t];
    out[(size_t)g * N_CLASSES + t] = o;
  }
}

extern "C" void kernel_launch(void* const* d_in, const int* in_sizes, int n_in,
                              void* d_out, int out_size, void* d_ws, size_t ws_size,
                              hipStream_t stream) {
  const float* x       = (const float*)d_in[0];
  const int*   ei      = (const int*)d_in[1];
  const int*   batch   = (const int*)d_in[2];
  const float* gin_w1  = (const float*)d_in[3];
  const float* gin_b1  = (const float*)d_in[4];
  const float* gin_w2  = (const float*)d_in[5];
  const float* gin_b2  = (const float*)d_in[6];
  const float* bn1_g   = (const float*)d_in[7];
  const float* bn1_b   = (const float*)d_in[8];
  const float* sage_wl = (const float*)d_in[9];
  const float* sage_bl = (const float*)d_in[10];
  const float* sage_wr = (const float*)d_in[11];
  const float* bn2_g   = (const float*)d_in[12];
  const float* bn2_b   = (const float*)d_in[13];
  const float* cls_w1  = (const float*)d_in[14];
  const float* cls_b1  = (const float*)d_in[15];
  const float* cls_w2  = (const float*)d_in[16];
  const float* cls_b2  = (const float*)d_in[17];
  float* out = (float*)d_out;
  float* ws  = (float*)d_ws;

  float* agg    = ws + OFF_AGG;
  float* deg    = ws + OFF_DEG;
  float* nbsum  = ws + OFF_NBSUM;
  float* gsum   = ws + OFF_GSUM;
  float* gcnt   = ws + OFF_GCNT;
  float* s1     = ws + OFF_S1;
  float* q1     = ws + OFF_Q1;
  float* s2     = ws + OFF_S2;
  float* q2     = ws + OFF_Q2;
  float* scale1 = ws + OFF_SCALE1;
  float* shift1 = ws + OFF_SHIFT1;
  float* scale2 = ws + OFF_SCALE2;
  float* shift2 = ws + OFF_SHIFT2;
  float* H      = ws + OFF_H;
  float* H3     = ws + OFF_H3;

  k_zero<<<(ZERO_COUNT + 255) / 256, 256, 0, stream>>>(ws, ZERO_COUNT);
  k_gin_agg<<<(N_EDGES + 255) / 256, 256, 0, stream>>>(x, ei, agg, deg);
  k_gin_mlp<<<(N_NODES + 63) / 64, 128, 0, stream>>>(x, agg, gin_w1, gin_b1,
                                                     gin_w2, gin_b2, H, s1, q1);
  k_bn_finalize<<<1, 64, 0, stream>>>(s1, q1, bn1_g, bn1_b, scale1, shift1);
  k_bn_relu<<<(N_NODES * HID + 255) / 256, 256, 0, stream>>>(H, scale1, shift1);
  k_sage_agg<<<(N_EDGES * 16 + 255) / 256, 256, 0, stream>>>(H, ei, nbsum);
  k_sage_lin<<<(N_NODES + 63) / 64, 128, 0, stream>>>(H, nbsum, deg, sage_wl,
                                                      sage_bl, sage_wr, H3, s2, q2);
  k_bn_finalize<<<1, 64, 0, stream>>>(s2, q2, bn2_g, bn2_b, scale2, shift2);
  k_bn2_pool<<<(N_NODES * 16 + 255) / 256, 256, 0, stream>>>(H3, scale2, shift2,
                                                             batch, gsum, gcnt);
  k_classifier<<<N_GRAPHS, 32, 0, stream>>>(gsum, gcnt, cls_w1, cls_b1,
                                            cls_w2, cls_b2, out);
}